// SimpleProcessor_21354577395753
// MI455X (gfx1250) — compile-verified
//
#include <hip/hip_runtime.h>
#include <hip/hip_bf16.h>

#define N_NODES 100000
#define N_EDGES 1600000

typedef __attribute__((ext_vector_type(2))) float v2f;
typedef __attribute__((ext_vector_type(8))) float v8f;

// ---------------- degree / normalization ----------------

__global__ void init_deg_kernel(float* __restrict__ deg, int n) {
    int i = blockIdx.x * blockDim.x + threadIdx.x;
    if (i < n) deg[i] = 1.0f;   // +1 for the implicit self loop
}

__global__ void deg_accum_kernel(const int* __restrict__ dst, float* __restrict__ deg, int e) {
    int i = blockIdx.x * blockDim.x + threadIdx.x;
    if (i < e) atomicAdd(&deg[dst[i]], 1.0f);
}

__global__ void finalize_dinv_kernel(float* __restrict__ deg, int n) {
    int i = blockIdx.x * blockDim.x + threadIdx.x;
    if (i < n) deg[i] = rsqrtf(deg[i]);   // in place: deg -> dinv
}

// ---------------- WMMA GEMM: h = x @ W^T ; agg = h*dinv^2 + b ----------------
// One wave computes a 16-row tile of nodes across all FOUT columns using
// V_WMMA_F32_16X16X4_F32. W ([FOUT][FIN] row-major) is staged in LDS.

template <int FIN, int FOUT>
__global__ __launch_bounds__(128) void gcn_gemm_kernel(
    const float* x,                    // [N, FIN]   (may alias agg region-wise: same rows only)
    const float* __restrict__ W,       // [FOUT, FIN]
    const float* __restrict__ bias,    // [FOUT]
    const float* __restrict__ dinv,    // [N]
    float* __restrict__ h,             // [N, FOUT]
    float* agg,                        // [N, FOUT]  seeded with self-loop term + bias
    int n_tiles)
{
    __shared__ float sW[FOUT * FIN];
    for (int i = threadIdx.x; i < FOUT * FIN; i += blockDim.x) sW[i] = W[i];
    __syncthreads();

    const int wave = threadIdx.x >> 5;
    const int lane = threadIdx.x & 31;
    const int tile = blockIdx.x * (blockDim.x >> 5) + wave;
    if (tile >= n_tiles) return;            // wave-uniform: EXEC stays all-ones

    const int row_base = tile * 16;
    const int arow = row_base + (lane & 15);
    const int koff = (lane >> 4) << 1;      // lanes 0-15 hold K pair {0,1}; 16-31 hold {2,3}
    const int ncol = lane & 15;

    constexpr int NT = FOUT / 16;
    v8f acc[NT] = {};

#pragma unroll
    for (int k0 = 0; k0 < FIN; k0 += 4) {
        v2f a;
        a.x = x[(size_t)arow * FIN + k0 + koff];
        a.y = x[(size_t)arow * FIN + k0 + koff + 1];
#pragma unroll
        for (int t = 0; t < NT; ++t) {
            const int colg = t * 16 + ncol;
            v2f bf;
            bf.x = sW[colg * FIN + k0 + koff];       // W^T[k][col] = W[col][k]
            bf.y = sW[colg * FIN + k0 + koff + 1];
            acc[t] = __builtin_amdgcn_wmma_f32_16x16x4_f32(
                false, a, false, bf, (short)0, acc[t], false, false);
        }
    }

    // C/D layout: vgpr i, lanes 0-15 -> M=i ; lanes 16-31 -> M=8+i ; N = lane&15
    const int mhi = (lane >> 4) * 8;
    float dn2[8];
#pragma unroll
    for (int i = 0; i < 8; ++i) {
        float dn = dinv[row_base + mhi + i];
        dn2[i] = dn * dn;                    // self-loop coefficient
    }

#pragma unroll
    for (int t = 0; t < NT; ++t) {
        const int colg = t * 16 + ncol;
        const float bv = bias[colg];
#pragma unroll
        for (int i = 0; i < 8; ++i) {
            const int m = row_base + mhi + i;
            const float hv = acc[t][i];
            h[(size_t)m * FOUT + colg]   = hv;
            agg[(size_t)m * FOUT + colg] = hv * dn2[i] + bv;
        }
    }
}

// ---------------- edge scatter: agg[dst] += h[src] * dinv[src]*dinv[dst] ----------------
// grid.x covers edges, grid.y covers F/4 feature chunks. h and edge_index are
// L2-resident (25.6MB + 12.8MB << 192MB L2), so gathers/atomics resolve in L2.

template <int F>
__global__ __launch_bounds__(256) void gcn_scatter_kernel(
    const int* __restrict__ src,
    const int* __restrict__ dst,
    const float* __restrict__ dinv,
    const float* __restrict__ h,
    float* __restrict__ agg,
    int n_edges)
{
    const int e = blockIdx.x * blockDim.x + threadIdx.x;
    if (e >= n_edges) return;
    const int s = src[e];
    const int d = dst[e];
    const float norm = dinv[s] * dinv[d];
    const int fg = blockIdx.y << 2;

    const float4 hv = *(const float4*)(h + (size_t)s * F + fg);
    float* ap = agg + (size_t)d * F + fg;
    atomicAdd(ap + 0, hv.x * norm);
    atomicAdd(ap + 1, hv.y * norm);
    atomicAdd(ap + 2, hv.z * norm);
    atomicAdd(ap + 3, hv.w * norm);
}

// ---------------- elementwise ReLU ----------------

__global__ void relu_kernel(const float* __restrict__ in, float* __restrict__ out, int n) {
    int i = blockIdx.x * blockDim.x + threadIdx.x;
    if (i < n) out[i] = fmaxf(in[i], 0.0f);
}

// ---------------- launch ----------------

extern "C" void kernel_launch(void* const* d_in, const int* in_sizes, int n_in,
                              void* d_out, int out_size, void* d_ws, size_t ws_size,
                              hipStream_t stream) {
    const float* x   = (const float*)d_in[0];          // [1, N, 32]
    const int*   ei  = (const int*)d_in[1];            // [2, E]
    const float* W1  = (const float*)d_in[2];
    const float* b1  = (const float*)d_in[3];
    const float* W2  = (const float*)d_in[4];
    const float* b2  = (const float*)d_in[5];
    const float* W3  = (const float*)d_in[6];
    const float* b3  = (const float*)d_in[7];
    const int* src = ei;
    const int* dst = ei + N_EDGES;

    char* ws = (char*)d_ws;
    float* dinv = (float*)ws;
    size_t off = (((size_t)N_NODES * 4) + 255) & ~(size_t)255;
    float* bufA = (float*)(ws + off);                  // [N, 64]
    off += (size_t)N_NODES * 64 * 4;
    float* bufB = (float*)(ws + off);                  // [N, 64]

    float* out = (float*)d_out;                        // [N, 32] fp32

    const int n_tiles = N_NODES / 16;                  // 6250, exact
    const int gemm_blocks = (n_tiles + 3) / 4;         // 4 waves/block
    const dim3 gemm_grid(gemm_blocks), gemm_block(128);
    const int eblk = (N_EDGES + 255) / 256;
    const int nthr = 256;

    // normalization coefficients
    init_deg_kernel<<<(N_NODES + 255) / 256, nthr, 0, stream>>>(dinv, N_NODES);
    deg_accum_kernel<<<eblk, nthr, 0, stream>>>(dst, dinv, N_EDGES);
    finalize_dinv_kernel<<<(N_NODES + 255) / 256, nthr, 0, stream>>>(dinv, N_NODES);

    // ---- layer 1: 32 -> 64 ----  x -> h in bufA, agg in bufB
    gcn_gemm_kernel<32, 64><<<gemm_grid, gemm_block, 0, stream>>>(
        x, W1, b1, dinv, bufA, bufB, n_tiles);
    gcn_scatter_kernel<64><<<dim3(eblk, 16), nthr, 0, stream>>>(
        src, dst, dinv, bufA, bufB, N_EDGES);
    relu_kernel<<<(N_NODES * 64 + 255) / 256, nthr, 0, stream>>>(bufB, bufB, N_NODES * 64);

    // ---- layer 2: 64 -> 64 ----  x=bufB -> h in bufA, agg back into bufB
    // (same-row read-before-write within each workgroup; no cross-tile overlap)
    gcn_gemm_kernel<64, 64><<<gemm_grid, gemm_block, 0, stream>>>(
        bufB, W2, b2, dinv, bufA, bufB, n_tiles);
    gcn_scatter_kernel<64><<<dim3(eblk, 16), nthr, 0, stream>>>(
        src, dst, dinv, bufA, bufB, N_EDGES);
    relu_kernel<<<(N_NODES * 64 + 255) / 256, nthr, 0, stream>>>(bufB, bufB, N_NODES * 64);

    // ---- layer 3: 64 -> 32 ----  x=bufB -> h in bufA[0:N*32), agg in bufA[N*32:N*64)
    float* h3   = bufA;
    float* agg3 = bufA + (size_t)N_NODES * 32;
    gcn_gemm_kernel<64, 32><<<gemm_grid, gemm_block, 0, stream>>>(
        bufB, W3, b3, dinv, h3, agg3, n_tiles);
    gcn_scatter_kernel<32><<<dim3(eblk, 8), nthr, 0, stream>>>(
        src, dst, dinv, h3, agg3, N_EDGES);
    relu_kernel<<<(N_NODES * 32 + 255) / 256, nthr, 0, stream>>>(agg3, out, N_NODES * 32);
}